// GATLayoutPredictor_72129680769339
// MI455X (gfx1250) — compile-verified
//
#include <hip/hip_runtime.h>
#include <hip/hip_bf16.h>

typedef __attribute__((ext_vector_type(2))) float v2f;
typedef __attribute__((ext_vector_type(8))) float v8f;

#define NODES 20000
#define HID_C 128
#define NEG_SLOPE 0.2f

// ---------------------------------------------------------------- fill
__global__ void fill_f32(float* __restrict__ p, float v, int n) {
    int i = blockIdx.x * blockDim.x + threadIdx.x;
    if (i < n) p[i] = v;
}

// ---------------------------------------------------------------- GEMM via f32 WMMA
// C[M,Ncol] = A[M,K] @ B[K,Ncol].  One wave per 16x64 output tile.
// Fragment layouts per CDNA5 ISA 7.12.2:
//   A 16x4 : lane<16 -> {v0,v1} = A[lane][k0+0..1]; lane>=16 -> A[lane-16][k0+2..3]
//   B 4x16 : lane<16 -> {v0,v1} = B[k0+0..1][lane]; lane>=16 -> B[k0+2..3][lane-16]
//   C 16x16: vgpr r  : lane<16 -> C[r][lane]; lane>=16 -> C[r+8][lane-16]
__global__ void gemm16x64_wmma(const float* __restrict__ A, const float* __restrict__ B,
                               float* __restrict__ C, int M, int K, int Ncol) {
    const int lane = threadIdx.x & 31;
    const int wave = (blockIdx.x * blockDim.x + threadIdx.x) >> 5;
    const int ntiles = Ncol >> 6;
    const int mt = wave / ntiles;
    const int nt = wave - mt * ntiles;
    if (mt >= (M >> 4)) return;

    const int row  = (mt << 4) + (lane & 15);
    const int kOff = (lane >> 4) << 1;            // 0 or 2
    const int col  = (nt << 6) + (lane & 15);
    const float* Arow = A + (size_t)row * K;

    v8f acc0 = {}, acc1 = {}, acc2 = {}, acc3 = {};
    for (int k0 = 0; k0 < K; k0 += 4) {
        const int ka = k0 + kOff;
        v2f a; a.x = Arow[ka]; a.y = Arow[ka + 1];
        const float* B0 = B + (size_t)ka * Ncol + col;
        const float* B1 = B0 + Ncol;
        v2f b0; b0.x = B0[0];  b0.y = B1[0];
        v2f b1; b1.x = B0[16]; b1.y = B1[16];
        v2f b2; b2.x = B0[32]; b2.y = B1[32];
        v2f b3; b3.x = B0[48]; b3.y = B1[48];
        acc0 = __builtin_amdgcn_wmma_f32_16x16x4_f32(false, a, false, b0, (short)0, acc0, false, false);
        acc1 = __builtin_amdgcn_wmma_f32_16x16x4_f32(false, a, false, b1, (short)0, acc1, false, false);
        acc2 = __builtin_amdgcn_wmma_f32_16x16x4_f32(false, a, false, b2, (short)0, acc2, false, false);
        acc3 = __builtin_amdgcn_wmma_f32_16x16x4_f32(false, a, false, b3, (short)0, acc3, false, false);
    }

    const int rowOut = (mt << 4) + ((lane >> 4) << 3);   // +0 or +8
    float* Cp = C + (size_t)rowOut * Ncol + (nt << 6) + (lane & 15);
#pragma unroll
    for (int r = 0; r < 8; ++r) {
        Cp[(size_t)r * Ncol + 0]  = acc0[r];
        Cp[(size_t)r * Ncol + 16] = acc1[r];
        Cp[(size_t)r * Ncol + 32] = acc2[r];
        Cp[(size_t)r * Ncol + 48] = acc3[r];
    }
}

// ---------------------------------------------------------------- per-(node,head) attention dots
__global__ void alpha_kernel(const float* __restrict__ h, const float* __restrict__ a_src,
                             const float* __restrict__ a_dst, float* __restrict__ asrc,
                             float* __restrict__ adst, int N, int heads) {
    const int wave = (blockIdx.x * blockDim.x + threadIdx.x) >> 5;
    const int lane = threadIdx.x & 31;
    if (wave >= N * heads) return;
    const int n = wave / heads, hd = wave - n * heads;
    const float4 hv = *(const float4*)(h + ((size_t)n * heads + hd) * HID_C + lane * 4);
    const float4 as = *(const float4*)(a_src + hd * HID_C + lane * 4);
    const float4 ad = *(const float4*)(a_dst + hd * HID_C + lane * 4);
    float ds = hv.x * as.x + hv.y * as.y + hv.z * as.z + hv.w * as.w;
    float dd = hv.x * ad.x + hv.y * ad.y + hv.z * ad.z + hv.w * ad.w;
    for (int off = 16; off > 0; off >>= 1) {
        ds += __shfl_down(ds, off, 32);
        dd += __shfl_down(dd, off, 32);
    }
    if (lane == 0) { asrc[wave] = ds; adst[wave] = dd; }
}

// ordered-uint encoding so atomicMax works on signed floats
__device__ __forceinline__ unsigned enc_f32(float f) {
    unsigned u = __float_as_uint(f);
    return (u & 0x80000000u) ? ~u : (u | 0x80000000u);
}
__device__ __forceinline__ float dec_f32(unsigned u) {
    return __uint_as_float((u & 0x80000000u) ? (u ^ 0x80000000u) : ~u);
}

__device__ __forceinline__ void edge_ids(const int* ei, int E, int e, int& s, int& d) {
    if (e < E) { s = ei[e]; d = ei[E + e]; }
    else       { s = e - E; d = e - E; }          // self loops appended
}

// ---------------------------------------------------------------- softmax pass 1: segment max
__global__ void edge_max_kernel(const int* __restrict__ ei, int E, int E2, int heads,
                                const float* __restrict__ asrc, const float* __restrict__ adst,
                                unsigned* __restrict__ mEnc) {
    int idx = blockIdx.x * blockDim.x + threadIdx.x;
    if (idx >= E2 * heads) return;
    int e = idx / heads, hd = idx - e * heads, s, d;
    edge_ids(ei, E, e, s, d);
    float v = asrc[s * heads + hd] + adst[d * heads + hd];
    v = v > 0.f ? v : NEG_SLOPE * v;
    atomicMax(mEnc + d * heads + hd, enc_f32(v));
}

// ---------------------------------------------------------------- softmax pass 2: exp + segment sum
__global__ void edge_expsum_kernel(const int* __restrict__ ei, int E, int E2, int heads,
                                   const float* __restrict__ asrc, const float* __restrict__ adst,
                                   const unsigned* __restrict__ mEnc, float* __restrict__ ssum,
                                   float* __restrict__ ealpha) {
    int idx = blockIdx.x * blockDim.x + threadIdx.x;
    if (idx >= E2 * heads) return;
    int e = idx / heads, hd = idx - e * heads, s, d;
    edge_ids(ei, E, e, s, d);
    float v = asrc[s * heads + hd] + adst[d * heads + hd];
    v = v > 0.f ? v : NEG_SLOPE * v;
    float e2 = __expf(v - dec_f32(mEnc[d * heads + hd]));
    ealpha[idx] = e2;
    atomicAdd(ssum + d * heads + hd, e2);
}

// ---------------------------------------------------------------- weighted scatter-aggregate
// one wave per (edge, head): 128 floats, 4 per lane, atomic f32 adds into L2-resident output
__global__ void edge_aggr_kernel(const int* __restrict__ ei, int E, int E2, int heads,
                                 const float* __restrict__ h, const float* __restrict__ ealpha,
                                 const float* __restrict__ ssum, float* __restrict__ out) {
    const int wave = (blockIdx.x * blockDim.x + threadIdx.x) >> 5;
    const int lane = threadIdx.x & 31;
    if (wave >= E2 * heads) return;
    int e = wave / heads, hd = wave - e * heads, s, d;
    edge_ids(ei, E, e, s, d);
    float w = ealpha[wave] / (ssum[d * heads + hd] + 1e-16f);
    const float4 hv = *(const float4*)(h + ((size_t)s * heads + hd) * HID_C + lane * 4);
    float* dp = out + ((size_t)d * heads + hd) * HID_C + lane * 4;
    atomicAdd(dp + 0, hv.x * w);
    atomicAdd(dp + 1, hv.y * w);
    atomicAdd(dp + 2, hv.z * w);
    atomicAdd(dp + 3, hv.w * w);
}

// ---------------------------------------------------------------- bias + optional ELU
__global__ void bias_act_kernel(float* __restrict__ buf, const float* __restrict__ bias,
                                int total, int rowlen, int doElu) {
    int i = blockIdx.x * blockDim.x + threadIdx.x;
    if (i >= total) return;
    float v = buf[i] + bias[i % rowlen];
    if (doElu) v = v > 0.f ? v : expm1f(v);
    buf[i] = v;
}

extern "C" void kernel_launch(void* const* d_in, const int* in_sizes, int n_in,
                              void* d_out, int out_size, void* d_ws, size_t ws_size,
                              hipStream_t stream) {
    const float* x   = (const float*)d_in[0];
    const int*   ei  = (const int*)d_in[1];
    const float* W1  = (const float*)d_in[2];
    const float* a1s = (const float*)d_in[3];
    const float* a1d = (const float*)d_in[4];
    const float* b1  = (const float*)d_in[5];
    const float* W2  = (const float*)d_in[6];
    const float* a2s = (const float*)d_in[7];
    const float* a2d = (const float*)d_in[8];
    const float* b2  = (const float*)d_in[9];
    const float* W3  = (const float*)d_in[10];
    const float* a3s = (const float*)d_in[11];
    const float* a3d = (const float*)d_in[12];
    const float* b3  = (const float*)d_in[13];

    const int N = NODES, HEADS = 4;
    const int E  = in_sizes[1] / 2;     // 400000
    const int E2 = E + N;               // + self loops
    const int F1 = HEADS * HID_C;       // 512

    char* ws = (char*)d_ws;
    auto alloc = [&](size_t bytes) -> char* {
        char* p = ws; ws += (bytes + 255) & ~(size_t)255; return p;
    };
    float*    buf0   = (float*)alloc((size_t)N * F1 * 4);
    float*    buf1   = (float*)alloc((size_t)N * F1 * 4);
    float*    asrc   = (float*)alloc((size_t)N * HEADS * 4);
    float*    adst   = (float*)alloc((size_t)N * HEADS * 4);
    unsigned* mEnc   = (unsigned*)alloc((size_t)N * HEADS * 4);
    float*    ssum   = (float*)alloc((size_t)N * HEADS * 4);
    float*    ealpha = (float*)alloc((size_t)E2 * HEADS * 4);

    auto cdiv = [](long long a, long long b) { return (int)((a + b - 1) / b); };

    auto run_layer = [&](const float* in, int K, const float* W, const float* as,
                         const float* ad, const float* bias, int heads,
                         float* hbuf, float* outbuf, int doElu) {
        const int Ncol = heads * HID_C;
        // GEMM: h = in @ W
        const int waves = (N / 16) * (Ncol / 64);
        gemm16x64_wmma<<<cdiv((long long)waves * 32, 256), 256, 0, stream>>>(in, W, hbuf, N, K, Ncol);
        // attention logits per (node, head)
        alpha_kernel<<<cdiv((long long)N * heads * 32, 256), 256, 0, stream>>>(hbuf, as, ad, asrc, adst, N, heads);
        // init softmax scratch + output accumulator (0-bits is the lowest mEnc key, below enc(-inf))
        fill_f32<<<cdiv(N * heads, 256), 256, 0, stream>>>((float*)mEnc, 0.f, N * heads);
        fill_f32<<<cdiv(N * heads, 256), 256, 0, stream>>>(ssum, 0.f, N * heads);
        fill_f32<<<cdiv((long long)N * Ncol, 256), 256, 0, stream>>>(outbuf, 0.f, N * Ncol);
        // segment softmax over incoming edges
        edge_max_kernel<<<cdiv((long long)E2 * heads, 256), 256, 0, stream>>>(ei, E, E2, heads, asrc, adst, mEnc);
        edge_expsum_kernel<<<cdiv((long long)E2 * heads, 256), 256, 0, stream>>>(ei, E, E2, heads, asrc, adst, mEnc, ssum, ealpha);
        // weighted aggregation
        edge_aggr_kernel<<<cdiv((long long)E2 * heads * 32, 256), 256, 0, stream>>>(ei, E, E2, heads, hbuf, ealpha, ssum, outbuf);
        // bias + activation
        bias_act_kernel<<<cdiv((long long)N * Ncol, 256), 256, 0, stream>>>(outbuf, bias, N * Ncol, Ncol, doElu);
    };

    // layer 1: x[N,768] -> buf1 = elu(gat(...)) [N,512]
    run_layer(x, 768, W1, a1s, a1d, b1, HEADS, buf0, buf1, 1);
    // layer 2: buf1 -> buf1 (GEMM writes buf0, aggregate back into buf1)
    run_layer(buf1, F1, W2, a2s, a2d, b2, HEADS, buf0, buf1, 1);
    // layer 3: buf1 -> d_out, 1 head, no ELU (mean over 1 head == identity)
    run_layer(buf1, F1, W3, a3s, a3d, b3, 1, buf0, (float*)d_out, 0);
}